// CAMD_67001489817945
// MI455X (gfx1250) — compile-verified
//
#include <hip/hip_runtime.h>

// CDNA5 / gfx1250. wave32. f32 WMMA 16x16x4.
typedef __attribute__((ext_vector_type(2))) float v2f;
typedef __attribute__((ext_vector_type(8))) float v8f;

// Explicit address-space pointers so codegen picks ds_* / global_* encodings
// even when the frontend pointer came through a runtime select.
typedef __attribute__((address_space(3))) float lds_float;
using gfloat_p = const float __attribute__((address_space(1))) *;
using gv2f_p   = const v2f   __attribute__((address_space(1))) *;

#define TT 4096
#define DD 64
#define LSTRIDE 66            // padded row stride (floats) to spread LDS banks
#define LBUF    (16 * LSTRIDE)

// One MLP layer for one 16-row tile, 2-way k-split WMMA ILP.
// 'a' points at this lane's A row (row m16), indexed by k.
template <bool RELU, typename AP, typename StoreFn>
__device__ __forceinline__ void mlp_layer(AP a, const float* __restrict__ Wl,
                                          const float* __restrict__ Bl,
                                          int m16, int khalf, StoreFn store)
{
#pragma unroll
    for (int nt = 0; nt < 4; ++nt) {
        const float bias = Bl[nt * 16 + m16];
        v8f c0, c1;
#pragma unroll
        for (int r = 0; r < 8; ++r) { c0[r] = bias; c1[r] = 0.0f; }

#pragma unroll
        for (int kb = 0; kb < DD; kb += 8) {
            v2f a0, a1, b0, b1;
            a0[0] = a[kb + khalf];         a0[1] = a[kb + khalf + 1];
            a1[0] = a[kb + 4 + khalf];     a1[1] = a[kb + 4 + khalf + 1];
            b0[0] = Wl[(kb + khalf)         * DD + nt * 16 + m16];
            b0[1] = Wl[(kb + khalf + 1)     * DD + nt * 16 + m16];
            b1[0] = Wl[(kb + 4 + khalf)     * DD + nt * 16 + m16];
            b1[1] = Wl[(kb + 4 + khalf + 1) * DD + nt * 16 + m16];
            c0 = __builtin_amdgcn_wmma_f32_16x16x4_f32(false, a0, false, b0,
                                                       (short)0, c0, false, false);
            c1 = __builtin_amdgcn_wmma_f32_16x16x4_f32(false, a1, false, b1,
                                                       (short)0, c1, false, false);
        }
#pragma unroll
        for (int r = 0; r < 8; ++r) {
            float v = c0[r] + c1[r];
            if (RELU) v = fmaxf(v, 0.0f);
            c0[r] = v;
        }
        store(nt, c0);
    }
}

// ---------------------------------------------------------------------------
// MLP kernel: Q (job 0, x1 with Wq) and K_m (jobs 1..4, x_m with Wk[m]) into
// ws[job] (4096x64 f32 row-major). One wave = one 16-row tile.
// ---------------------------------------------------------------------------
__global__ __launch_bounds__(128) void CAMD_mlp_kernel(
    const float* __restrict__ x1, const float* __restrict__ x2,
    const float* __restrict__ x3, const float* __restrict__ x4,
    const float* __restrict__ Wq_w, const float* __restrict__ Wq_b,
    const float* __restrict__ Wk_w, const float* __restrict__ Wk_b,
    float* __restrict__ ws)
{
    __shared__ float lds_storage[4][2 * LBUF];   // per-wave ping-pong

    const int lane  = threadIdx.x & 31;
    const int wave  = threadIdx.x >> 5;
    const int tile  = blockIdx.x * 4 + wave;   // 0..255
    const int job   = blockIdx.y;              // 0..4
    const int i0    = tile * 16;

    const int m16   = lane & 15;
    const int khalf = (lane >> 4) * 2;         // 0 or 2
    const int rbase = (lane >> 4) << 3;        // 0 or 8

    lds_float* ldsw = (lds_float*)&lds_storage[wave][0];

    const float* xs[5] = { x1, x1, x2, x3, x4 };
    gfloat_p x = (gfloat_p)xs[job];            // force global addressing
    const float* W = (job == 0) ? Wq_w : (Wk_w + (size_t)(job - 1) * 3 * DD * DD);
    const float* B = (job == 0) ? Wq_b : (Wk_b + (size_t)(job - 1) * 3 * DD);
    float* out = ws + (size_t)job * TT * DD;

    // ---- layer 0: A from global x, write LDS buf0 ----
    mlp_layer<true>(
        x + (size_t)(i0 + m16) * DD,
        W, B, m16, khalf,
        [&](int nt, const v8f& c) {
#pragma unroll
            for (int r = 0; r < 8; ++r)
                ldsw[(rbase + r) * LSTRIDE + nt * 16 + m16] = c[r];
        });
    __syncthreads();

    // ---- layer 1: LDS buf0 -> LDS buf1 ----
    mlp_layer<true>(
        (const lds_float*)(ldsw + m16 * LSTRIDE),
        W + 1 * DD * DD, B + 1 * DD, m16, khalf,
        [&](int nt, const v8f& c) {
#pragma unroll
            for (int r = 0; r < 8; ++r)
                ldsw[LBUF + (rbase + r) * LSTRIDE + nt * 16 + m16] = c[r];
        });
    __syncthreads();

    // ---- layer 2: LDS buf1 -> global (no ReLU) ----
    mlp_layer<false>(
        (const lds_float*)(ldsw + LBUF + m16 * LSTRIDE),
        W + 2 * DD * DD, B + 2 * DD, m16, khalf,
        [&](int nt, const v8f& c) {
#pragma unroll
            for (int r = 0; r < 8; ++r)
                out[(size_t)(i0 + rbase + r) * DD + nt * 16 + m16] = c[r];
        });
}

// ---------------------------------------------------------------------------
// Attention: one wave = one 16-row Q tile for one stream m.
// out[i,0:2] += sum_j (Q[i]·K_m[j]) * (t2[j] <= t1[i]) * V_m[j,0:2]
// Two j-tiles (32 cols) per iteration, k-split -> 4 independent WMMA chains.
// ---------------------------------------------------------------------------
__global__ __launch_bounds__(128) void CAMD_attn_kernel(
    const float* __restrict__ x1, const float* __restrict__ x2,
    const float* __restrict__ x3, const float* __restrict__ x4,
    const float* __restrict__ ws, float* __restrict__ out)
{
    const int lane  = threadIdx.x & 31;
    const int wave  = threadIdx.x >> 5;
    const int tile  = blockIdx.x * 4 + wave;   // 0..255
    const int m     = blockIdx.y;              // 0..3
    const int i0    = tile * 16;

    const int m16   = lane & 15;
    const int khalf = (lane >> 4) * 2;
    const int rbase = (lane >> 4) << 3;

    const float* xs[4] = { x1, x2, x3, x4 };
    const float* xmg = xs[m];                  // generic copy (prefetch only)
    gfloat_p xm = (gfloat_p)xmg;               // force global addressing
    const float* Q  = ws;
    const float* K  = ws + (size_t)(1 + m) * TT * DD;

    // Register-resident Q fragments (cover K-dim 0..63); 8B-aligned v2f loads.
    v2f qf[16];
#pragma unroll
    for (int kk = 0; kk < 16; ++kk)
        qf[kk] = *(const v2f*)(Q + (size_t)(i0 + m16) * DD + kk * 4 + khalf);

    float t1v[8];
#pragma unroll
    for (int r = 0; r < 8; ++r)
        t1v[r] = x1[(size_t)(i0 + rbase + r) * DD + (DD - 1)];
    const float tmax = x1[(size_t)(i0 + 15) * DD + (DD - 1)];   // t1 sorted

    float acc0[8], acc1[8];
#pragma unroll
    for (int r = 0; r < 8; ++r) { acc0[r] = 0.0f; acc1[r] = 0.0f; }

    for (int j0 = 0; j0 < TT; j0 += 32) {
        if (xm[(size_t)j0 * DD + (DD - 1)] > tmax) break;   // uniform early exit

        const int jA = j0 + m16;
        const int jB = j0 + 16 + m16;

        // Prefetch next j-tile (K rows + V/t rows) behind this tile's WMMAs.
        if (j0 + 32 < TT) {
            __builtin_prefetch(K   + (size_t)(jA + 32) * DD, 0, 1);
            __builtin_prefetch(K   + (size_t)(jB + 32) * DD, 0, 1);
            __builtin_prefetch(xmg + (size_t)(jA + 32) * DD, 0, 1);
        }

        const float t2a = xm[(size_t)jA * DD + (DD - 1)];
        const v2f   va  = *(gv2f_p)(xm + (size_t)jA * DD);
        const float t2b = xm[(size_t)jB * DD + (DD - 1)];
        const v2f   vb  = *(gv2f_p)(xm + (size_t)jB * DD);

        v8f ca0 = {}, ca1 = {}, cb0 = {}, cb1 = {};
#pragma unroll
        for (int kk = 0; kk < 16; kk += 2) {
            const float* kpa = K + (size_t)jA * DD + kk * 4 + khalf;
            const float* kpb = K + (size_t)jB * DD + kk * 4 + khalf;
            v2f ba0 = *(const v2f*)(kpa);
            v2f ba1 = *(const v2f*)(kpa + 4);
            v2f bb0 = *(const v2f*)(kpb);
            v2f bb1 = *(const v2f*)(kpb + 4);
            ca0 = __builtin_amdgcn_wmma_f32_16x16x4_f32(false, qf[kk],     false, ba0,
                                                        (short)0, ca0, false, false);
            cb0 = __builtin_amdgcn_wmma_f32_16x16x4_f32(false, qf[kk],     false, bb0,
                                                        (short)0, cb0, false, false);
            ca1 = __builtin_amdgcn_wmma_f32_16x16x4_f32(false, qf[kk + 1], false, ba1,
                                                        (short)0, ca1, false, false);
            cb1 = __builtin_amdgcn_wmma_f32_16x16x4_f32(false, qf[kk + 1], false, bb1,
                                                        (short)0, cb1, false, false);
        }

        // c[r] = S[i0+rbase+r][jX]; mask + fold in V[:,0:2].
#pragma unroll
        for (int r = 0; r < 8; ++r) {
            float sa = ca0[r] + ca1[r];
            sa = (t2a <= t1v[r]) ? sa : 0.0f;
            acc0[r] += sa * va[0];
            acc1[r] += sa * va[1];
            float sb = cb0[r] + cb1[r];
            sb = (t2b <= t1v[r]) ? sb : 0.0f;
            acc0[r] += sb * vb[0];
            acc1[r] += sb * vb[1];
        }
    }

    // Reduce over the 16 lanes of each half-wave (S-tile columns).
#pragma unroll
    for (int off = 8; off >= 1; off >>= 1) {
#pragma unroll
        for (int r = 0; r < 8; ++r) {
            acc0[r] += __shfl_xor(acc0[r], off, 32);
            acc1[r] += __shfl_xor(acc1[r], off, 32);
        }
    }

    if (m16 == 0) {   // lane 0 -> rows i0..i0+7, lane 16 -> rows i0+8..i0+15
#pragma unroll
        for (int r = 0; r < 8; ++r) {
            atomicAdd(&out[(size_t)(i0 + rbase + r) * 2 + 0], acc0[r]);
            atomicAdd(&out[(size_t)(i0 + rbase + r) * 2 + 1], acc1[r]);
        }
    }
}

extern "C" void kernel_launch(void* const* d_in, const int* in_sizes, int n_in,
                              void* d_out, int out_size, void* d_ws, size_t ws_size,
                              hipStream_t stream) {
    const float* x1   = (const float*)d_in[0];
    const float* x2   = (const float*)d_in[1];
    const float* x3   = (const float*)d_in[2];
    const float* x4   = (const float*)d_in[3];
    const float* Wq_w = (const float*)d_in[4];
    const float* Wq_b = (const float*)d_in[5];
    const float* Wk_w = (const float*)d_in[6];
    const float* Wk_b = (const float*)d_in[7];

    float* ws  = (float*)d_ws;    // 5 * 4096 * 64 f32 = 5.25 MB (Q, K0..K3)
    float* out = (float*)d_out;   // 4096 * 2 f32

    hipMemsetAsync(d_out, 0, (size_t)out_size * sizeof(float), stream);

    dim3 mgrid(64, 5);
    CAMD_mlp_kernel<<<mgrid, 128, 0, stream>>>(x1, x2, x3, x4,
                                               Wq_w, Wq_b, Wk_w, Wk_b, ws);

    dim3 agrid(64, 4);
    CAMD_attn_kernel<<<agrid, 128, 0, stream>>>(x1, x2, x3, x4, ws, out);
}